// DistanceMinimizingGNN_63170378989884
// MI455X (gfx1250) — compile-verified
//
#include <hip/hip_runtime.h>
#include <hip/hip_bf16.h>

typedef __attribute__((ext_vector_type(2))) float v2f;
typedef __attribute__((ext_vector_type(8))) float v8f;

#define Nn   1024
#define Hd   128
#define Ee   32768

// ---------------------------------------------------------------------------
// Degree / normalization kernels
// ---------------------------------------------------------------------------
__global__ void k_init_deg(float* deg) {
    int i = blockIdx.x * blockDim.x + threadIdx.x;
    if (i < Nn) deg[i] = 1.0f;               // self-loop contributes 1
}

__global__ void k_count_deg(const int* __restrict__ ei, float* deg) {
    int e = blockIdx.x * blockDim.x + threadIdx.x;
    if (e < Ee) atomicAdd(&deg[ei[Ee + e]], 1.0f);   // dst = ei[1][e]
}

__global__ void k_deg_to_dinv(float* deg) {
    int i = blockIdx.x * blockDim.x + threadIdx.x;
    if (i < Nn) deg[i] = rsqrtf(deg[i]);     // deg >= 1 always
}

__global__ void k_zero(float* p, int n) {
    int i = blockIdx.x * blockDim.x + threadIdx.x;
    if (i < n) p[i] = 0.0f;
}

// ---------------------------------------------------------------------------
// fp32 WMMA GEMM: C[1024 x 128] = A[1024 x 128] * B[128 x 128] (+bias)
// B[k][n] = Bp[k*BSK + n*BSN]  (BSK=128,BSN=1 normal; BSK=1,BSN=256 for dp_w1^T)
// One wave per 16x16 tile via V_WMMA_F32_16X16X4_F32; 8 waves/block -> 16x128 strip.
// Compile-time strides + bias flag -> branch-free epilogue, immediate-offset
// addressing for fragment loads and stores.
// ---------------------------------------------------------------------------
template <int BSK, int BSN, bool ADD_BIAS>
__global__ void __launch_bounds__(256)
k_gemm_wmma(const float* __restrict__ A, const float* __restrict__ Bp,
            float* __restrict__ C, const float* __restrict__ bias) {
    const int lane  = threadIdx.x & 31;
    const int wave  = threadIdx.x >> 5;
    const int m0    = blockIdx.x << 4;       // 16-row strip
    const int n0    = wave << 4;             // 16-col tile per wave
    const int lhalf = lane >> 4;             // 0/1 -> K-half select
    const int lmod  = lane & 15;

    // Fragment base pointers; K advances by constant strides.
    const float* ap = A + (m0 + lmod) * 128 + (lhalf << 1);   // A: lane%16 = M
    const float* bp = Bp + (lhalf << 1) * BSK + (n0 + lmod) * BSN; // B: lane%16 = N

    v8f acc = {};
#pragma unroll 8
    for (int k0 = 0; k0 < 128; k0 += 4) {
        v2f a, b;
        a.x = ap[k0];                        // contiguous pair -> b64
        a.y = ap[k0 + 1];
        b.x = bp[k0 * BSK];                  // BSK=1: contiguous pair -> b64
        b.y = bp[(k0 + 1) * BSK];            // BSK=128: two b32, imm offsets
        // 8 args: (neg_a, A, neg_b, B, c_mod, C, reuse_a, reuse_b)
        acc = __builtin_amdgcn_wmma_f32_16x16x4_f32(
            false, a, false, b, (short)0, acc, false, false);
    }

    // D layout: M = r + 8*(lane/16), N = lane%16. Bias invariant across rows.
    float bv = 0.0f;
    if (ADD_BIAS) bv = bias[n0 + lmod];
    float* cp = C + (m0 + (lhalf << 3)) * 128 + n0 + lmod;
#pragma unroll
    for (int r = 0; r < 8; ++r) {
        float val = acc[r];
        if (ADD_BIAS) val += bv;
        cp[r * 128] = val;                   // constant 512B stride -> imm offsets
    }
}

// ---------------------------------------------------------------------------
// Edge aggregation: agg[dst] += xw[src] * dinv[src]*dinv[dst]  (edges + self loops)
// 32 threads per edge, float4 per thread -> 128 features.
// ---------------------------------------------------------------------------
__global__ void k_edge_agg(const int* __restrict__ ei, const float* __restrict__ xw,
                           const float* __restrict__ dinv, float* __restrict__ agg) {
    const int t    = blockIdx.x * blockDim.x + threadIdx.x;
    const int edge = t >> 5;
    const int f    = (t & 31) << 2;
    if (edge >= Ee + Nn) return;
    int s, d;
    if (edge < Ee) { s = ei[edge]; d = ei[Ee + edge]; }
    else           { s = d = edge - Ee; }    // self loop
    const float norm = dinv[s] * dinv[d];
    const float4 xv  = *(const float4*)(xw + s * 128 + f);
    float* dst = agg + d * 128 + f;
    atomicAdd(dst + 0, xv.x * norm);
    atomicAdd(dst + 1, xv.y * norm);
    atomicAdd(dst + 2, xv.z * norm);
    atomicAdd(dst + 3, xv.w * norm);
}

// h = (relu?) (agg + bias)
template <bool RELU>
__global__ void k_post_agg(const float* __restrict__ agg, const float* __restrict__ bias,
                           float* __restrict__ h) {
    int i = blockIdx.x * blockDim.x + threadIdx.x;
    if (i >= Nn * 128) return;
    float v = agg[i] + bias[i & 127];
    if (RELU) v = fmaxf(v, 0.0f);
    h[i] = v;
}

// ---------------------------------------------------------------------------
// Pairwise head: p(i,j) = sigmoid( sum_h relu(u[i][h]+v[j][h]+b1[h]) * w2[h] + b2 )
// for i<j; dist[i][j] = dist[j][i] = p; diagonal = 0. 16x16 pair tile per block,
// u/v tiles + b1/w2 staged in LDS (17KB of 320KB WGP pool).
// ---------------------------------------------------------------------------
__global__ void __launch_bounds__(256)
k_pairwise(const float* __restrict__ u, const float* __restrict__ v,
           const float* __restrict__ b1, const float* __restrict__ w2,
           const float* __restrict__ b2p, float* __restrict__ dist) {
    const int ti = blockIdx.y, tj = blockIdx.x;
    if (ti > tj) return;                     // covered by mirror writes of (tj,ti)

    __shared__ float su[16][128];
    __shared__ float sv[16][128];
    __shared__ float sb[128];
    __shared__ float sw[128];

    const int tid = threadIdx.x;
    for (int k = tid; k < 16 * 128; k += 256) {
        const int r = k >> 7, c = k & 127;
        su[r][c] = u[(ti * 16 + r) * 128 + c];
        sv[r][c] = v[(tj * 16 + r) * 128 + c];
    }
    if (tid < 128) { sb[tid] = b1[tid]; sw[tid] = w2[tid]; }
    __syncthreads();

    const int li = tid >> 4, lj = tid & 15;
    const int i = ti * 16 + li, j = tj * 16 + lj;

    float acc = 0.0f;
#pragma unroll 4
    for (int h = 0; h < 128; ++h) {
        const float t = su[li][h] + sv[lj][h] + sb[h];
        acc = fmaf(fmaxf(t, 0.0f), sw[h], acc);
    }

    if (i < j) {
        const float p = 1.0f / (1.0f + __expf(-(acc + b2p[0])));
        dist[i * Nn + j] = p;
        dist[j * Nn + i] = p;
    } else if (i == j) {
        dist[i * Nn + i] = 0.0f;
    }
}

// ---------------------------------------------------------------------------
extern "C" void kernel_launch(void* const* d_in, const int* in_sizes, int n_in,
                              void* d_out, int out_size, void* d_ws, size_t ws_size,
                              hipStream_t stream) {
    const float* x     = (const float*)d_in[0];
    const int*   ei    = (const int*)  d_in[1];
    const float* convW[3] = { (const float*)d_in[2], (const float*)d_in[4], (const float*)d_in[6] };
    const float* convB[3] = { (const float*)d_in[3], (const float*)d_in[5], (const float*)d_in[7] };
    const float* out_w = (const float*)d_in[8];
    const float* out_b = (const float*)d_in[9];
    const float* dp_w1 = (const float*)d_in[10];   // [128, 256] row-major
    const float* dp_b1 = (const float*)d_in[11];
    const float* dp_w2 = (const float*)d_in[12];   // [1, 128]
    const float* dp_b2 = (const float*)d_in[13];   // [1]

    float* emb  = (float*)d_out;                   // [1024*128]
    float* dist = emb + Nn * Hd;                   // [1024*1024]

    float* ws   = (float*)d_ws;
    float* deg  = ws;                              // 1024 (becomes dinv in place)
    float* xw   = deg + 1024;                      // 131072
    float* agg  = xw  + Nn * Hd;                   // 131072
    float* h    = agg + Nn * Hd;                   // 131072
    float* ub   = h   + Nn * Hd;                   // 131072
    float* vb   = ub  + Nn * Hd;                   // 131072

    // degree -> dinv
    k_init_deg   <<<4,   256, 0, stream>>>(deg);
    k_count_deg  <<<128, 256, 0, stream>>>(ei, deg);
    k_deg_to_dinv<<<4,   256, 0, stream>>>(deg);

    // 3 GCN conv layers
    const float* cur = x;
    for (int l = 0; l < 3; ++l) {
        k_gemm_wmma<128, 1, false><<<64, 256, 0, stream>>>(cur, convW[l], xw, nullptr);
        k_zero     <<<512, 256, 0, stream>>>(agg, Nn * Hd);
        k_edge_agg <<<4224, 256, 0, stream>>>(ei, xw, deg, agg);
        if (l < 2) k_post_agg<true ><<<512, 256, 0, stream>>>(agg, convB[l], h);
        else       k_post_agg<false><<<512, 256, 0, stream>>>(agg, convB[l], h);
        cur = h;
    }

    // node embeddings = h @ out_w + out_b  -> directly into d_out
    k_gemm_wmma<128, 1, true><<<64, 256, 0, stream>>>(h, out_w, emb, out_b);

    // u = emb @ dp_w1[:, :H]^T ; v = emb @ dp_w1[:, H:]^T  (B[k][n] = dp_w1[n*256+k])
    k_gemm_wmma<1, 256, false><<<64, 256, 0, stream>>>(emb, dp_w1,       ub, nullptr);
    k_gemm_wmma<1, 256, false><<<64, 256, 0, stream>>>(emb, dp_w1 + 128, vb, nullptr);

    // pairwise distances
    dim3 pg(Nn / 16, Nn / 16);
    k_pairwise<<<pg, 256, 0, stream>>>(ub, vb, dp_b1, dp_w2, dp_b2, dist);
}